// LightGCN_90460601188414
// MI455X (gfx1250) — compile-verified
//
#include <hip/hip_runtime.h>
#include <hip/hip_bf16.h>
#include <stdint.h>

#define NNODES  200000
#define EMB_D   128
#define EPB     256      // edges staged per block

#define AS1 __attribute__((address_space(1)))
#define AS3 __attribute__((address_space(3)))

#if defined(__has_builtin)
#  if __has_builtin(__builtin_amdgcn_global_load_async_to_lds_b32)
#    define USE_ASYNC_BUILTIN 1
#  endif
#  if __has_builtin(__builtin_amdgcn_s_wait_asynccnt)
#    define USE_WAITASYNC_BUILTIN 1
#  endif
#endif

typedef __attribute__((ext_vector_type(2))) float v2f;
typedef __attribute__((ext_vector_type(8))) float v8f;

// ---------------------------------------------------------------- async stage
__device__ __forceinline__ void stage_b32_to_lds(const int* gsrc, int* lds_dst) {
#if defined(USE_ASYNC_BUILTIN)
  __builtin_amdgcn_global_load_async_to_lds_b32(
      (AS1 int*)(uintptr_t)gsrc,
      (AS3 int*)(uint32_t)(uintptr_t)lds_dst, 0, 0);
#else
  asm volatile("global_load_async_to_lds_b32 %0, %1, off"
               :
               : "v"((uint32_t)(uintptr_t)lds_dst), "v"(gsrc)
               : "memory");
#endif
}

__device__ __forceinline__ void wait_async_lds() {
#if defined(USE_WAITASYNC_BUILTIN)
  __builtin_amdgcn_s_wait_asynccnt(0);
#else
  asm volatile("s_wait_asynccnt 0" ::: "memory");
#endif
}

// ---------------------------------------------------------------- utilities
__global__ void lgcn_zero4(float4* __restrict__ p, int n4) {
  int i = blockIdx.x * blockDim.x + threadIdx.x;
  int stride = gridDim.x * blockDim.x;
  float4 z = {0.f, 0.f, 0.f, 0.f};
  for (; i < n4; i += stride) p[i] = z;
}

__global__ void lgcn_add4(float4* __restrict__ dst, const float4* __restrict__ src, int n4) {
  int i = blockIdx.x * blockDim.x + threadIdx.x;
  int stride = gridDim.x * blockDim.x;
  for (; i < n4; i += stride) {
    float4 a = dst[i];
    float4 b = src[i];
    a.x += b.x; a.y += b.y; a.z += b.z; a.w += b.w;
    dst[i] = a;
  }
}

__global__ void lgcn_deg(const int* __restrict__ col, float* __restrict__ deg, int E) {
  int e = blockIdx.x * blockDim.x + threadIdx.x;
  if (e < E) unsafeAtomicAdd(&deg[col[e]], 1.0f);
}

__global__ void lgcn_dis(float* __restrict__ deg, int n) {
  int i = blockIdx.x * blockDim.x + threadIdx.x;
  if (i < n) {
    float d = deg[i];
    deg[i] = (d > 0.f) ? rsqrtf(d) : 0.f;   // d>=1 when >0, matches max(deg,1)
  }
}

// ---------------------------------------------------------------- conv layer
// One wave per edge iteration: 32 lanes x float4 = 128-float row.
// Edge indices staged to LDS via gfx1250 async global->LDS path.
__global__ __launch_bounds__(256) void lgcn_conv(
    const float* __restrict__ x,
    const int* __restrict__ row,
    const int* __restrict__ col,
    const float* __restrict__ dis,
    float* __restrict__ out,
    int num_edges)
{
  __shared__ int   s_row[EPB];
  __shared__ int   s_col[EPB];
  __shared__ float s_norm[EPB];

  const int t    = threadIdx.x;
  const int base = blockIdx.x * EPB;
  int e = base + t;
  if (e >= num_edges) e = num_edges - 1;   // clamp; tail entries never consumed

  stage_b32_to_lds(row + e, &s_row[t]);
  stage_b32_to_lds(col + e, &s_col[t]);
  wait_async_lds();
  __syncthreads();

  s_norm[t] = dis[s_row[t]] * dis[s_col[t]];
  __syncthreads();

  const int wid  = t >> 5;
  const int lane = t & 31;
  const int coff = lane * 4;

#pragma unroll 1
  for (int i = 0; i < 32; ++i) {
    const int le = (wid << 5) + i;
    if (base + le >= num_edges) break;     // uniform within wave
    const int   r  = s_row[le];
    const int   c  = s_col[le];
    const float nm = s_norm[le];
    const float4 xv = *(const float4*)(x + (size_t)r * EMB_D + coff);
    float* op = out + (size_t)c * EMB_D + coff;
    unsafeAtomicAdd(op + 0, nm * xv.x);
    unsafeAtomicAdd(op + 1, nm * xv.y);
    unsafeAtomicAdd(op + 2, nm * xv.z);
    unsafeAtomicAdd(op + 3, nm * xv.w);
  }
}

// ---------------------------------------------------------------- finalize
// out = (out + add) * 0.25, done per 16x16 tile as X * (0.25*I16) via four
// chained v_wmma_f32_16x16x4_f32 (numerically exact). One wave per tile.
__global__ __launch_bounds__(256) void lgcn_finalize_wmma(
    float* __restrict__ out, const float* __restrict__ add)
{
  const int wave     = (blockIdx.x * blockDim.x + threadIdx.x) >> 5;
  const int lane     = threadIdx.x & 31;
  const int tile_row = wave >> 3;            // NNODES/16 row tiles
  const int tile_col = wave & 7;             // 128/16 col tiles
  const size_t r0 = (size_t)tile_row * 16;
  const int    c0 = tile_col * 16;
  const int    m     = lane & 15;            // A: M index / B,D: N index
  const int    khalf = (lane >> 4) * 2;      // hi half-wave covers K=2,3

  const size_t rowoff = (r0 + m) * EMB_D;
  v8f acc = {};
#pragma unroll
  for (int c = 0; c < 4; ++c) {
    const int kcol = c0 + c * 4 + khalf;     // A_c = X[:, 4c..4c+3]
    v2f a, b;
    a.x = out[rowoff + kcol]     + add[rowoff + kcol];
    a.y = out[rowoff + kcol + 1] + add[rowoff + kcol + 1];
    // B_c = 0.25 * I16[4c..4c+3, :]  (lane holds N=m, VGPR v holds K=khalf+v)
    b.x = (m == c * 4 + khalf)     ? 0.25f : 0.0f;
    b.y = (m == c * 4 + khalf + 1) ? 0.25f : 0.0f;
    acc = __builtin_amdgcn_wmma_f32_16x16x4_f32(
        false, a, false, b, (short)0, acc, false, false);
  }

  // D layout: lanes 0-15 -> M=v, N=lane; lanes 16-31 -> M=8+v, N=lane-16
  const int mbase = (lane >> 4) * 8;
#pragma unroll
  for (int v = 0; v < 8; ++v) {
    out[(r0 + mbase + v) * EMB_D + c0 + m] = acc[v];
  }
}

// ---------------------------------------------------------------- launcher
extern "C" void kernel_launch(void* const* d_in, const int* in_sizes, int n_in,
                              void* d_out, int out_size, void* d_ws, size_t ws_size,
                              hipStream_t stream) {
  const int E = in_sizes[0] / 2;                     // edge_index is [2, E]
  const int N = in_sizes[1] / EMB_D;                 // NNODES

  const int*   edge = (const int*)d_in[0];
  const float* emb  = (const float*)d_in[1];
  const int*   row  = edge;                          // edge_index[0]
  const int*   col  = edge + E;                      // edge_index[1]
  float*       out  = (float*)d_out;

  // workspace layout: deg (N floats, padded to 128) | bufA | bufB
  float* deg  = (float*)d_ws;
  size_t degPad = ((size_t)N + 127) & ~(size_t)127;
  float* bufA = deg + degPad;
  float* bufB = bufA + (size_t)N * EMB_D;

  const int    nFeat  = N * EMB_D;
  const int    nFeat4 = nFeat / 4;
  const size_t featBytes = (size_t)nFeat * sizeof(float);

  const int TPB = 256;
  const int zeroBlocks = 4096;
  const int convBlocks = (E + EPB - 1) / EPB;
  const int tileBlocks = (N / 16) * (EMB_D / 16) / (TPB / 32);  // one wave/tile

  // degrees -> deg_inv_sqrt
  lgcn_zero4<<<256, TPB, 0, stream>>>((float4*)deg, N / 4);
  lgcn_deg<<<(E + TPB - 1) / TPB, TPB, 0, stream>>>(col, deg, E);
  lgcn_dis<<<(N + TPB - 1) / TPB, TPB, 0, stream>>>(deg, N);

  // acc = emb (layer 0 term)
  (void)hipMemcpyAsync(out, emb, featBytes, hipMemcpyDeviceToDevice, stream);

  // layer 1: emb -> bufA ; acc += bufA
  lgcn_zero4<<<zeroBlocks, TPB, 0, stream>>>((float4*)bufA, nFeat4);
  lgcn_conv<<<convBlocks, TPB, 0, stream>>>(emb, row, col, deg, bufA, E);
  lgcn_add4<<<zeroBlocks, TPB, 0, stream>>>((float4*)out, (const float4*)bufA, nFeat4);

  // layer 2: bufA -> bufB ; acc += bufB
  lgcn_zero4<<<zeroBlocks, TPB, 0, stream>>>((float4*)bufB, nFeat4);
  lgcn_conv<<<convBlocks, TPB, 0, stream>>>(bufA, row, col, deg, bufB, E);
  lgcn_add4<<<zeroBlocks, TPB, 0, stream>>>((float4*)out, (const float4*)bufB, nFeat4);

  // layer 3: bufB -> bufA ; acc = (acc + bufA) * 0.25 via WMMA
  lgcn_zero4<<<zeroBlocks, TPB, 0, stream>>>((float4*)bufA, nFeat4);
  lgcn_conv<<<convBlocks, TPB, 0, stream>>>(bufB, row, col, deg, bufA, E);
  lgcn_finalize_wmma<<<tileBlocks, TPB, 0, stream>>>(out, bufA);
}